// AttentionPooling_49108656062866
// MI455X (gfx1250) — compile-verified
//
#include <hip/hip_runtime.h>
#include <math.h>

typedef __attribute__((ext_vector_type(2))) float v2f;
typedef __attribute__((ext_vector_type(8))) float v8f;

static constexpr int kN   = 524288;   // nodes
static constexpr int kD   = 256;      // features
static constexpr int kSeg = 4096;     // segments
static constexpr int kBlocksA  = kN / 128;  // 4096 (8 waves x 16 rows per block)
static constexpr int kBlocksB2 = kN / 256;  // 2048

// ---------------------------------------------------------------------------
// Kernel A: scores[i] = dot(X[i,:], W) + b via V_WMMA_F32_16X16X4_F32.
// Each wave computes a 16-row tile with 64 WMMAs (K=4 each, f32 exact).
// A-frag (ISA 7.12.2, 32-bit A 16x4): lanes 0-15 hold M=lane with {K0,K1} in
// {v0,v1}; lanes 16-31 hold the same rows with {K2,K3}.  -> one b64 load/lane.
// B-frag: broadcast the W chunk to ALL lanes so every output column of C is
// the same score (immunizes us against B's column<->lane mapping).
// D-frag: lane 0 holds scores for M=0..7 in c[0..7], lane 16 for M=8..15.
// ---------------------------------------------------------------------------
__global__ void __launch_bounds__(256)
scores_wmma_kernel(const float* __restrict__ X, const float* __restrict__ W,
                   const float* __restrict__ Bb, float* __restrict__ scores,
                   float* __restrict__ blockmax) {
  __shared__ float smax[16];
  const int lane = threadIdx.x & 31;
  const int wave = threadIdx.x >> 5;
  const int m15  = lane & 15;
  const int half = lane >> 4;                       // 0: K0/K1, 1: K2/K3
  const int rowTile = blockIdx.x * 8 + wave;        // 16 rows per tile

  const float* rowp = X + (size_t)(rowTile * 16 + m15) * kD + (half << 1);
  v8f c = {};

#pragma unroll 4
  for (int k = 0; k < kD / 4; ++k) {
    v2f a = *(const v2f*)(rowp + k * 4);                  // 16B of row per pair
    v2f b = *(const v2f*)(W + k * 4 + (half << 1));       // broadcast W chunk
    c = __builtin_amdgcn_wmma_f32_16x16x4_f32(false, a, false, b,
                                              (short)0, c, false, false);
  }

  if (m15 == 0) {                                   // lanes 0 and 16 hold N=0
    const float bias  = Bb[0];
    const int   mbase = rowTile * 16 + (half << 3);
    float m = -INFINITY;
#pragma unroll
    for (int r = 0; r < 8; ++r) {
      float s = c[r] + bias;
      scores[mbase + r] = s;
      m = fmaxf(m, s);
    }
    smax[wave * 2 + half] = m;
  }
  __syncthreads();
  if (threadIdx.x == 0) {
    float m = smax[0];
#pragma unroll
    for (int i = 1; i < 16; ++i) m = fmaxf(m, smax[i]);
    blockmax[blockIdx.x] = m;
  }
}

// ---------------------------------------------------------------------------
// Kernel B1: global max of the 4096 per-block maxima (order-independent).
// ---------------------------------------------------------------------------
__global__ void __launch_bounds__(1024)
reduce_max_kernel(const float* __restrict__ blockmax, float* __restrict__ gmax) {
  __shared__ float sm[1024];
  float m = -INFINITY;
  for (int i = threadIdx.x; i < kBlocksA; i += 1024) m = fmaxf(m, blockmax[i]);
  sm[threadIdx.x] = m;
  __syncthreads();
  for (int s = 512; s > 0; s >>= 1) {
    if (threadIdx.x < s) sm[threadIdx.x] = fmaxf(sm[threadIdx.x], sm[threadIdx.x + s]);
    __syncthreads();
  }
  if (threadIdx.x == 0) gmax[0] = sm[0];
}

// ---------------------------------------------------------------------------
// Kernel B2: scores[i] <- exp(scores[i]-gmax); fixed-order partial sums.
// ---------------------------------------------------------------------------
__global__ void __launch_bounds__(256)
exp_partial_kernel(float* __restrict__ scores, const float* __restrict__ gmax,
                   float* __restrict__ blocksum) {
  __shared__ float ss[256];
  const int i = blockIdx.x * 256 + threadIdx.x;
  const float m = gmax[0];
  const float e = expf(scores[i] - m);
  scores[i] = e;
  ss[threadIdx.x] = e;
  __syncthreads();
  for (int s = 128; s > 0; s >>= 1) {
    if (threadIdx.x < s) ss[threadIdx.x] += ss[threadIdx.x + s];
    __syncthreads();
  }
  if (threadIdx.x == 0) blocksum[blockIdx.x] = ss[0];
}

// ---------------------------------------------------------------------------
// Kernel B3: deterministic tree sum of 2048 partials -> inv = 1/sum.
// ---------------------------------------------------------------------------
__global__ void __launch_bounds__(1024)
reduce_sum_kernel(const float* __restrict__ blocksum, float* __restrict__ invp) {
  __shared__ float ss[1024];
  ss[threadIdx.x] = blocksum[threadIdx.x] + blocksum[threadIdx.x + 1024];
  __syncthreads();
  for (int s = 512; s > 0; s >>= 1) {
    if (threadIdx.x < s) ss[threadIdx.x] += ss[threadIdx.x + s];
    __syncthreads();
  }
  if (threadIdx.x == 0) invp[0] = 1.0f / ss[0];
}

// ---------------------------------------------------------------------------
// Kernel C: one block per segment; batch_index is sorted -> uniform binary
// search for [start,end).  256 threads = 256 feature columns, fully coalesced
// streaming FMA over the segment's contiguous rows; single store, no atomics.
// ---------------------------------------------------------------------------
__global__ void __launch_bounds__(256)
segment_pool_kernel(const float* __restrict__ X, const int* __restrict__ idx,
                    const float* __restrict__ escores,
                    const float* __restrict__ invp, float* __restrict__ out) {
  const int seg = blockIdx.x;
  const int t   = threadIdx.x;

  int lo = 0, hi = kN;                       // lower_bound(seg)  (uniform)
  while (lo < hi) { int mid = (lo + hi) >> 1; if (idx[mid] < seg) lo = mid + 1; else hi = mid; }
  const int start = lo;
  hi = kN;                                   // lower_bound(seg+1), lo>=start
  while (lo < hi) { int mid = (lo + hi) >> 1; if (idx[mid] < seg + 1) lo = mid + 1; else hi = mid; }
  const int end = lo;

  const float inv = invp[0];
  float acc = 0.0f;
  int i = start;
  for (; i + 4 <= end; i += 4) {
    const int pf = (i + 16 <= kN - 1) ? i + 16 : kN - 1;   // stream-ahead hint
    __builtin_prefetch(&X[(size_t)pf * kD + t], 0, 3);
    const float a0 = escores[i + 0] * inv;
    const float a1 = escores[i + 1] * inv;
    const float a2 = escores[i + 2] * inv;
    const float a3 = escores[i + 3] * inv;
    acc = fmaf(X[(size_t)(i + 0) * kD + t], a0, acc);
    acc = fmaf(X[(size_t)(i + 1) * kD + t], a1, acc);
    acc = fmaf(X[(size_t)(i + 2) * kD + t], a2, acc);
    acc = fmaf(X[(size_t)(i + 3) * kD + t], a3, acc);
  }
  for (; i < end; ++i)
    acc = fmaf(X[(size_t)i * kD + t], escores[i] * inv, acc);

  out[(size_t)seg * kD + t] = acc;
}

// ---------------------------------------------------------------------------
// Launch.  Inputs: [0] node_features f32[N*D], [1] batch_index i32[N],
// [2] num_segments (scalar), [3] W f32[D], [4] b f32[1].  Output f32[4096*256].
// Workspace (floats): [0,N) scores/exp-scores | [N,N+4096) blockmax |
// [N+4096,N+6144) blocksum | N+6144: gmax | N+6145: inv.  (~2.03 MB)
// ---------------------------------------------------------------------------
extern "C" void kernel_launch(void* const* d_in, const int* in_sizes, int n_in,
                              void* d_out, int out_size, void* d_ws, size_t ws_size,
                              hipStream_t stream) {
  const float* X   = (const float*)d_in[0];
  const int*   idx = (const int*)d_in[1];
  const float* W   = (const float*)d_in[3];
  const float* Bb  = (const float*)d_in[4];
  float*       out = (float*)d_out;

  float* ws       = (float*)d_ws;
  float* scores   = ws;
  float* blockmax = ws + kN;
  float* blocksum = ws + kN + kBlocksA;
  float* gmax     = ws + kN + kBlocksA + kBlocksB2;
  float* invp     = gmax + 1;

  scores_wmma_kernel<<<kBlocksA, 256, 0, stream>>>(X, W, Bb, scores, blockmax);
  reduce_max_kernel<<<1, 1024, 0, stream>>>(blockmax, gmax);
  exp_partial_kernel<<<kBlocksB2, 256, 0, stream>>>(scores, gmax, blocksum);
  reduce_sum_kernel<<<1, 1024, 0, stream>>>(blocksum, invp);
  segment_pool_kernel<<<kSeg, 256, 0, stream>>>(X, idx, scores, invp, out);
}